// LinOSSLayer_52621939310911
// MI455X (gfx1250) — compile-verified
//
#include <hip/hip_runtime.h>
#include <hip/hip_bf16.h>
#include <math.h>

// Problem constants (from reference): L=16384, H=256, S=512
#define LSEQ 16384
#define HDIM 256
#define SDIM 512
#define NCHUNK 128
#define CLEN 128   // NCHUNK * CLEN == LSEQ

typedef float v2f __attribute__((ext_vector_type(2)));
typedef float v4f __attribute__((ext_vector_type(4)));
typedef float v8f __attribute__((ext_vector_type(8)));
typedef int   v4i __attribute__((ext_vector_type(4)));

__device__ __forceinline__ v8f wmma_f32(v2f a, v2f b, v8f c) {
    // D = A(16x4 f32) * B(4x16 f32) + C(16x16 f32)
    return __builtin_amdgcn_wmma_f32_16x16x4_f32(
        /*neg_a=*/false, a, /*neg_b=*/false, b,
        /*c_mod=*/(short)0, c, /*reuse_a=*/false, /*reuse_b=*/false);
}

// ---------------------------------------------------------------------------
// Async global->LDS staging (gfx1250 ASYNCcnt path), with sync fallback.
// Builtin signature (from compiler diagnostic): b128 flavor takes typed
// pointers: (int4 AS(1)*, int4 AS(3)*, imm offset, imm cpol).
// ---------------------------------------------------------------------------
#if __has_builtin(__builtin_amdgcn_global_load_async_to_lds_b128) && \
    __has_builtin(__builtin_amdgcn_s_wait_asynccnt)
#define HAVE_ASYNC_LDS 1
#else
#define HAVE_ASYNC_LDS 0
#warning "gfx1250 async-to-LDS builtins unavailable; using synchronous LDS staging"
#endif

typedef __attribute__((address_space(1))) v4i* g_v4i_ptr;
typedef __attribute__((address_space(3))) v4i* l_v4i_ptr;

__device__ __forceinline__ void cp16_g2l(const float* __restrict__ g, float* l) {
#if HAVE_ASYNC_LDS
    // Generic LDS pointers carry the LDS offset in their low 32 bits, so the
    // integer round-trip to an AS(3) pointer is value-correct.
    __builtin_amdgcn_global_load_async_to_lds_b128(
        (g_v4i_ptr)(uintptr_t)g,
        (l_v4i_ptr)(unsigned)(uintptr_t)l,
        /*offset=*/0, /*cpol=*/0);
#else
    *(v4f*)l = *(const v4f*)g;
#endif
}

__device__ __forceinline__ void async_wait0() {
#if HAVE_ASYNC_LDS
    __builtin_amdgcn_s_wait_asynccnt(0);
#endif
}
__device__ __forceinline__ void async_wait4() {
#if HAVE_ASYNC_LDS
    __builtin_amdgcn_s_wait_asynccnt(4);
#endif
}

// ---------------------------------------------------------------------------
// GEMM 1: Bu = x @ B_c^T   (L,H)x(H,S) -> (L,S) complex (re & im planes)
// Block: 256 threads = 8 waves, tile 32(L) x 64(S). X tile async-staged in LDS.
// ---------------------------------------------------------------------------
__global__ __launch_bounds__(256) void gemm1_kernel(
    const float* __restrict__ X,   // (L,H)
    const float* __restrict__ Bm,  // (S,H,2)
    float* __restrict__ BuRe,      // (L,S)
    float* __restrict__ BuIm)      // (L,S)
{
    __shared__ float lx[32][HDIM + 4];   // +4 pad: bank stride 4 mod 64; row=1040B (16B-aligned)

    const int tid  = threadIdx.x;
    const int row0 = blockIdx.x * 32;    // over L
    const int col0 = blockIdx.y * 64;    // over S

    // async coalesced stage of the 32xH X tile: 2048 16B units / 256 threads
    #pragma unroll
    for (int u = 0; u < 8; ++u) {
        const int unit = tid + u * 256;          // 0..2047
        const int r    = unit >> 6;              // / 64 units per row
        const int c    = (unit & 63) * 4;
        cp16_g2l(&X[(size_t)(row0 + r) * HDIM + c], &lx[r][c]);
    }
    async_wait0();
    __syncthreads();

    const int wave = tid >> 5;
    const int lane = tid & 31;
    const int wr = wave >> 2;            // 0..1  row sub-tile
    const int wc = wave & 3;             // 0..3  col sub-tile
    const int m  = lane & 15;            // M index for A, N index for B
    const int hi = lane >> 4;            // K-half selector

    v8f accR = {};
    v8f accI = {};

    const int scol = col0 + wc * 16 + m; // output column (state channel)
    const float* __restrict__ brow = Bm + (size_t)scol * (HDIM * 2);
    const float* __restrict__ arow = &lx[wr * 16 + m][0];

    for (int kb = 0; kb < HDIM; kb += 4) {
        const int k = kb + 2 * hi;
        v2f a;  a[0] = arow[k];          a[1] = arow[k + 1];
        v2f br; br[0] = brow[2 * k];     br[1] = brow[2 * (k + 1)];
        v2f bi; bi[0] = brow[2 * k + 1]; bi[1] = brow[2 * (k + 1) + 1];
        accR = wmma_f32(a, br, accR);
        accI = wmma_f32(a, bi, accI);
    }

    for (int r = 0; r < 8; ++r) {
        const int mrow = row0 + wr * 16 + r + 8 * hi;
        BuRe[(size_t)mrow * SDIM + scol] = accR[r];
        BuIm[(size_t)mrow * SDIM + scol] = accI[r];
    }
}

// ---------------------------------------------------------------------------
// Per-channel transition constants: M11,M12,M21,M22, f1=M11*step, f2=M21*step
// ---------------------------------------------------------------------------
__global__ void precompute_kernel(const float* __restrict__ A_diag,
                                  const float* __restrict__ steps,
                                  float* __restrict__ mc)  // (S,6)
{
    const int s = blockIdx.x * blockDim.x + threadIdx.x;
    if (s >= SDIM) return;
    const float A  = fmaxf(A_diag[s], 0.0f);
    const float st = 1.0f / (1.0f + expf(-steps[s]));
    const float schur = 1.0f / (1.0f + st * st * A);
    const float M11 = 1.0f - st * st * A * schur;
    const float M12 = -st * A * schur;
    const float M21 = st * schur;
    const float M22 = schur;
    float* p = mc + (size_t)s * 6;
    p[0] = M11; p[1] = M12; p[2] = M21; p[3] = M22;
    p[4] = M11 * st; p[5] = M21 * st;
}

// ---------------------------------------------------------------------------
// Scan phase A: chunk-local states. grid (S/256, NCHUNK), block 256.
// Coalesced across channels (innermost dim of Bu).
// ---------------------------------------------------------------------------
__global__ __launch_bounds__(256) void scan_local_kernel(
    const float* __restrict__ BuRe, const float* __restrict__ BuIm,
    const float* __restrict__ mc, float* __restrict__ carry) // (NCHUNK,S,4)
{
    const int s = blockIdx.x * 256 + threadIdx.x;
    const int j = blockIdx.y;
    const float* p = mc + (size_t)s * 6;
    const float M11 = p[0], M12 = p[1], M21 = p[2], M22 = p[3];
    const float f1 = p[4], f2 = p[5];

    float b1r = 0.f, b1i = 0.f, b2r = 0.f, b2i = 0.f;
    size_t base = (size_t)j * CLEN * SDIM + s;
    for (int t = 0; t < CLEN; ++t) {
        const float ur = BuRe[base];
        const float ui = BuIm[base];
        const float n1r = M11 * b1r + M12 * b2r + f1 * ur;
        const float n1i = M11 * b1i + M12 * b2i + f1 * ui;
        const float n2r = M21 * b1r + M22 * b2r + f2 * ur;
        const float n2i = M21 * b1i + M22 * b2i + f2 * ui;
        b1r = n1r; b1i = n1i; b2r = n2r; b2i = n2i;
        base += SDIM;
    }
    float* c = carry + ((size_t)j * SDIM + s) * 4;
    c[0] = b1r; c[1] = b1i; c[2] = b2r; c[3] = b2i;
}

// ---------------------------------------------------------------------------
// Scan phase B: scan the NCHUNK chunk carries per channel. P = M^CLEN via
// 7 squarings of the 2x2. Writes the incoming state for each chunk.
// ---------------------------------------------------------------------------
__global__ __launch_bounds__(256) void scan_carry_kernel(
    const float* __restrict__ carry, const float* __restrict__ mc,
    float* __restrict__ prefix)  // (NCHUNK,S,4)
{
    const int s = blockIdx.x * blockDim.x + threadIdx.x;
    if (s >= SDIM) return;
    const float* p = mc + (size_t)s * 6;
    float P11 = p[0], P12 = p[1], P21 = p[2], P22 = p[3];
    #pragma unroll
    for (int i = 0; i < 7; ++i) {        // M^(2^7) = M^128 = M^CLEN
        const float a = P11 * P11 + P12 * P21;
        const float b = P11 * P12 + P12 * P22;
        const float c = P21 * P11 + P22 * P21;
        const float d = P21 * P12 + P22 * P22;
        P11 = a; P12 = b; P21 = c; P22 = d;
    }
    float c1r = 0.f, c1i = 0.f, c2r = 0.f, c2i = 0.f;
    for (int j = 0; j < NCHUNK; ++j) {
        float* pf = prefix + ((size_t)j * SDIM + s) * 4;
        pf[0] = c1r; pf[1] = c1i; pf[2] = c2r; pf[3] = c2i;
        const float* cb = carry + ((size_t)j * SDIM + s) * 4;
        const float n1r = P11 * c1r + P12 * c2r + cb[0];
        const float n1i = P11 * c1i + P12 * c2i + cb[1];
        const float n2r = P21 * c1r + P22 * c2r + cb[2];
        const float n2i = P21 * c1i + P22 * c2i + cb[3];
        c1r = n1r; c1i = n1i; c2r = n2r; c2i = n2i;
    }
}

// ---------------------------------------------------------------------------
// Scan phase C: replay each chunk with the correct incoming state; overwrite
// Bu in place with ys (second state component, complex).
// ---------------------------------------------------------------------------
__global__ __launch_bounds__(256) void scan_replay_kernel(
    float* __restrict__ BuRe, float* __restrict__ BuIm,
    const float* __restrict__ mc, const float* __restrict__ prefix)
{
    const int s = blockIdx.x * 256 + threadIdx.x;
    const int j = blockIdx.y;
    const float* p = mc + (size_t)s * 6;
    const float M11 = p[0], M12 = p[1], M21 = p[2], M22 = p[3];
    const float f1 = p[4], f2 = p[5];

    const float* pf = prefix + ((size_t)j * SDIM + s) * 4;
    float b1r = pf[0], b1i = pf[1], b2r = pf[2], b2i = pf[3];

    size_t base = (size_t)j * CLEN * SDIM + s;
    for (int t = 0; t < CLEN; ++t) {
        const float ur = BuRe[base];
        const float ui = BuIm[base];
        const float n1r = M11 * b1r + M12 * b2r + f1 * ur;
        const float n1i = M11 * b1i + M12 * b2i + f1 * ui;
        const float n2r = M21 * b1r + M22 * b2r + f2 * ur;
        const float n2i = M21 * b1i + M22 * b2i + f2 * ui;
        b1r = n1r; b1i = n1i; b2r = n2r; b2i = n2i;
        BuRe[base] = b2r;               // ys real
        BuIm[base] = b2i;               // ys imag
        base += SDIM;
    }
}

// ---------------------------------------------------------------------------
// GEMM 2: out = Re(ys @ C_c^T) + D*x = ysRe@CreT - ysIm@CimT + D*x
// (L,S)x(S,H) -> (L,H). ys tiles double-buffered through LDS via async copies
// in K-chunks of 64 (4 async b128 per thread per chunk -> s_wait_asynccnt 4).
// ---------------------------------------------------------------------------
#define KC2 64
#define NKC2 (SDIM / KC2)   // 8 chunks

__global__ __launch_bounds__(256) void gemm2_kernel(
    const float* __restrict__ YRe, const float* __restrict__ YIm,
    const float* __restrict__ Cm,   // (H,S,2)
    const float* __restrict__ X,    // (L,H)
    const float* __restrict__ D,    // (H)
    float* __restrict__ Out)        // (L,H)
{
    // [buf][row][col]; row stride 68 floats = 272B (16B aligned, 4 banks mod 64)
    __shared__ float lre[2][32][KC2 + 4];
    __shared__ float lim[2][32][KC2 + 4];

    const int tid  = threadIdx.x;
    const int row0 = blockIdx.x * 32;    // over L
    const int col0 = blockIdx.y * 64;    // over H

    const int wave = tid >> 5;
    const int lane = tid & 31;
    const int wr = wave >> 2;
    const int wc = wave & 3;
    const int m  = lane & 15;
    const int hi = lane >> 4;

    const int hcol = col0 + wc * 16 + m;
    const float* __restrict__ crow = Cm + (size_t)hcol * (SDIM * 2);

    // per-thread staging coords: 512 16B units per plane per chunk, 256 threads
    const int u0r = (tid + 0)   >> 4;           // unit/ (KC2/4=16 units per row)
    const int u0c = ((tid + 0)  & 15) * 4;
    const int u1r = (tid + 256) >> 4;
    const int u1c = ((tid + 256) & 15) * 4;

    auto prefetch = [&](int kc, int buf) {
        cp16_g2l(&YRe[(size_t)(row0 + u0r) * SDIM + kc + u0c], &lre[buf][u0r][u0c]);
        cp16_g2l(&YRe[(size_t)(row0 + u1r) * SDIM + kc + u1c], &lre[buf][u1r][u1c]);
        cp16_g2l(&YIm[(size_t)(row0 + u0r) * SDIM + kc + u0c], &lim[buf][u0r][u0c]);
        cp16_g2l(&YIm[(size_t)(row0 + u1r) * SDIM + kc + u1c], &lim[buf][u1r][u1c]);
    };

    v8f acc = {};

    prefetch(0, 0);
    for (int c = 0; c < NKC2; ++c) {
        const int buf = c & 1;
        if (c + 1 < NKC2) {
            prefetch((c + 1) * KC2, buf ^ 1);
            async_wait4();               // chunk c landed (in-order), c+1 in flight
        } else {
            async_wait0();
        }
        __syncthreads();                 // all waves' chunk-c data visible

        const float* __restrict__ arr = &lre[buf][wr * 16 + m][0];
        const float* __restrict__ ari = &lim[buf][wr * 16 + m][0];
        const int kc = c * KC2;
        for (int kb = 0; kb < KC2; kb += 4) {
            const int k  = kb + 2 * hi;
            const int kg = kc + k;
            v2f ar; ar[0] = arr[k]; ar[1] = arr[k + 1];
            v2f ai; ai[0] = ari[k]; ai[1] = ari[k + 1];
            v2f br; br[0] = crow[2 * kg];      br[1] = crow[2 * (kg + 1)];
            // fp32 WMMA has no A/B NEG -> negate imag B fragment at load
            v2f bi; bi[0] = -crow[2 * kg + 1]; bi[1] = -crow[2 * (kg + 1) + 1];
            acc = wmma_f32(ar, br, acc);
            acc = wmma_f32(ai, bi, acc);
        }
        __syncthreads();                 // done reading buf before it is re-targeted
    }

    const float d = D[hcol];
    for (int r = 0; r < 8; ++r) {
        const int mrow = row0 + wr * 16 + r + 8 * hi;
        const size_t o = (size_t)mrow * HDIM + hcol;
        Out[o] = acc[r] + d * X[o];
    }
}

// ---------------------------------------------------------------------------
extern "C" void kernel_launch(void* const* d_in, const int* in_sizes, int n_in,
                              void* d_out, int out_size, void* d_ws, size_t ws_size,
                              hipStream_t stream) {
    const float* X      = (const float*)d_in[0];  // (L,H)
    const float* A_diag = (const float*)d_in[1];  // (S)
    const float* Bm     = (const float*)d_in[2];  // (S,H,2)
    const float* Cm     = (const float*)d_in[3];  // (H,S,2)
    const float* Dv     = (const float*)d_in[4];  // (H)
    const float* steps  = (const float*)d_in[5];  // (S)
    float* Out = (float*)d_out;                   // (L,H)

    float* ws     = (float*)d_ws;
    float* BuRe   = ws;                                   // L*S
    float* BuIm   = BuRe   + (size_t)LSEQ * SDIM;         // L*S
    float* carry  = BuIm   + (size_t)LSEQ * SDIM;         // NCHUNK*S*4
    float* prefix = carry  + (size_t)NCHUNK * SDIM * 4;   // NCHUNK*S*4
    float* mc     = prefix + (size_t)NCHUNK * SDIM * 4;   // S*6

    // 1) Bu = x @ B_c^T (complex)
    {
        dim3 grid(LSEQ / 32, SDIM / 64);
        gemm1_kernel<<<grid, 256, 0, stream>>>(X, Bm, BuRe, BuIm);
    }
    // 2) per-channel constants
    precompute_kernel<<<2, 256, 0, stream>>>(A_diag, steps, mc);
    // 3) chunked scan: local -> carry scan -> replay (ys overwrites Bu)
    {
        dim3 grid(SDIM / 256, NCHUNK);
        scan_local_kernel<<<grid, 256, 0, stream>>>(BuRe, BuIm, mc, carry);
        scan_carry_kernel<<<2, 256, 0, stream>>>(carry, mc, prefix);
        scan_replay_kernel<<<grid, 256, 0, stream>>>(BuRe, BuIm, mc, prefix);
    }
    // 4) out = Re(ys @ C_c^T) + D*x
    {
        dim3 grid(LSEQ / 32, HDIM / 64);
        gemm2_kernel<<<grid, 256, 0, stream>>>(BuRe, BuIm, Cm, X, Dv, Out);
    }
}